// ExodusNet_2199023255636
// MI455X (gfx1250) — compile-verified
//
#include <hip/hip_runtime.h>

typedef __attribute__((ext_vector_type(2))) float v2f;
typedef __attribute__((ext_vector_type(8))) float v8f;

#define ALPHA           0.9048374180359595f   // exp(-1/10)
#define ONE_MINUS_ALPHA 0.0951625819640405f
#define THR             1.0f
#define T_STEPS         100
#define FEATS           32

// ---------------------------------------------------------------------------
// Phase 1: weighted[b,t] = sum_f w[f] * x[b,f,t]  via V_WMMA_F32_16X16X4_F32.
// One wave (32 lanes) per batch b. A-fragment: w broadcast to all 16 rows
// (so every row of D equals the wanted 1x16 result). B-fragment: x tile,
// n = t (lane&15), k = vgpr + 2*(lane>=16)  -- same k-mapping as the
// documented A layout. 8 chained K=4 WMMAs cover f = 0..31.
// ---------------------------------------------------------------------------
__global__ __launch_bounds__(256) void snn_gemv_wmma(const float* __restrict__ x,
                                                     const float* __restrict__ w,
                                                     float* __restrict__ weighted,
                                                     int nB)
{
    const int lane = threadIdx.x & 31;
    const int wave = threadIdx.x >> 5;
    const int b    = blockIdx.x * 8 + wave;
    if (b >= nB) return;

    const int hi = lane >> 4;   // 0: lanes 0-15, 1: lanes 16-31
    const int lo = lane & 15;

    // Build the 8 A-fragments once: A[m,k] = w[4*kb + k] for every row m.
    // Per ISA layout: VGPR0 holds k = 2*hi, VGPR1 holds k = 2*hi + 1.
    float wlane = (lane < FEATS) ? w[lane] : 0.0f;
    v2f a[8];
#pragma unroll
    for (int kb = 0; kb < 8; ++kb) {
        int f0 = kb * 4 + hi * 2;
        a[kb].x = __shfl(wlane, f0,     32);
        a[kb].y = __shfl(wlane, f0 + 1, 32);
    }

    const float* xb = x + (size_t)b * (FEATS * T_STEPS);

#pragma unroll 1
    for (int c = 0; c < 7; ++c) {             // 7 tiles of 16 timesteps (96 + 4)
        const int t0 = c * 16;
        int t = t0 + lo;
        if (t > T_STEPS - 1) t = T_STEPS - 1; // clamp; clamped columns discarded

        v8f d = {};
#pragma unroll
        for (int kb = 0; kb < 8; ++kb) {
            const int f0 = kb * 4 + hi * 2;   // k-mapping mirrors A fragment
            v2f bf;
            bf.x = xb[(size_t)f0       * T_STEPS + t];  // lanes 0-15: 64B coalesced
            bf.y = xb[(size_t)(f0 + 1) * T_STEPS + t];
            d = __builtin_amdgcn_wmma_f32_16x16x4_f32(
                    /*neg_a=*/false, a[kb],
                    /*neg_b=*/false, bf,
                    /*c_mod=*/(short)0, d,
                    /*reuse_a=*/false, /*reuse_b=*/false);
        }

        // All D rows are identical; lane n (lanes 0-15, VGPR0, m=0) holds
        // weighted[b, t0+n].
        if (hi == 0 && (t0 + lo) < T_STEPS)
            weighted[(size_t)b * T_STEPS + t0 + lo] = d[0];
    }
}

// ---------------------------------------------------------------------------
// Phase 2: fused ExpLeak + LIF scans, one thread per batch.
//   syn[t] = a*syn[t-1] + i[t]
//   v[t]   = a*v[t-1] + (1-a)*syn[t];  s = (v >= 1);  v -= s
// Reads 100 floats / writes 100 floats contiguously (float4).
// Safe when `weighted` aliases `out`: each element is read before it is
// overwritten by the same thread (data dependency through syn/v).
// ---------------------------------------------------------------------------
__global__ __launch_bounds__(256) void snn_scan(const float* weighted,
                                                float* out,
                                                int nB)
{
    const int b = blockIdx.x * blockDim.x + threadIdx.x;
    if (b >= nB) return;

    const float4* wp = (const float4*)(weighted + (size_t)b * T_STEPS);
    float4*       op = (float4*)(out + (size_t)b * T_STEPS);

    float syn = 0.0f, v = 0.0f;
#pragma unroll 1
    for (int tc = 0; tc < T_STEPS / 4; ++tc) {
        float4 iv = wp[tc];
        float  ii[4] = {iv.x, iv.y, iv.z, iv.w};
        float  ss[4];
#pragma unroll
        for (int j = 0; j < 4; ++j) {
            syn = ALPHA * syn + ii[j];
            v   = ALPHA * v + ONE_MINUS_ALPHA * syn;
            float s = (v >= THR) ? 1.0f : 0.0f;
            v -= s;                 // MembraneSubtract
            ss[j] = s;
        }
        float4 sv;
        sv.x = ss[0]; sv.y = ss[1]; sv.z = ss[2]; sv.w = ss[3];
        op[tc] = sv;
    }
}

// ---------------------------------------------------------------------------
extern "C" void kernel_launch(void* const* d_in, const int* in_sizes, int n_in,
                              void* d_out, int out_size, void* d_ws, size_t ws_size,
                              hipStream_t stream) {
    const float* x = (const float*)d_in[0];   // (B, 2, 4, 4, 100) = (B, 32, 100)
    const float* w = (const float*)d_in[1];   // (1, 32)
    float* out = (float*)d_out;               // (B, 1, 100)

    const int nB = in_sizes[0] / (FEATS * T_STEPS);   // 32768
    const size_t interBytes = (size_t)nB * T_STEPS * sizeof(float);

    // Intermediate currents: use scratch if big enough, else in-place via d_out.
    float* wbuf = (ws_size >= interBytes) ? (float*)d_ws : out;

    const int wavesPerBlock = 8;  // 256 threads = 8 wave32
    dim3 g1((nB + wavesPerBlock - 1) / wavesPerBlock);
    snn_gemv_wmma<<<g1, 256, 0, stream>>>(x, w, wbuf, nB);

    dim3 g2((nB + 255) / 256);
    snn_scan<<<g2, 256, 0, stream>>>(wbuf, out, nB);
}